// AdjointOperator_19731079758153
// MI455X (gfx1250) — compile-verified
//
#include <hip/hip_runtime.h>
#include <hip/hip_bf16.h>
#include <stdint.h>

#define DET 512
#define NANG 720
#define KA 8                 // angles per staged block
#define NBLK (NANG / KA)     // 90
#define TILE_X 64
#define TILE_Y 64
#define THREADS 256
#define RPT 16               // rows (pixels) per thread = TILE_Y / (THREADS/TILE_X)

#if defined(__has_builtin)
#if __has_builtin(__builtin_amdgcn_tensor_load_to_lds) && __has_builtin(__builtin_amdgcn_s_wait_tensorcnt)
#define USE_TDM 1
#endif
#endif
#ifndef USE_TDM
#define USE_TDM 0
#endif

typedef unsigned int u32x4 __attribute__((ext_vector_type(4)));
typedef int i32x4 __attribute__((ext_vector_type(4)));
typedef int i32x8 __attribute__((ext_vector_type(8)));

#if USE_TDM
// Issue a TDM DMA of an 8x512 f32 tile (one angle block, one batch) into LDS.
// D# per CDNA5 ISA ch.8: group0 = {count/flags, lds_addr, global_addr lo,
// global_addr hi | type=2}; group1 = dims/strides with data_size = 4B.
// This toolchain's builtin is the 6-arg form:
//   (u32x4 g0, i32x8 g1, i32x4 g2, i32x4 g3, i32x8 gx, i32 cpol)
// Groups 2/3 (and the trailing group) are unused for a 2-D tile (tile_dim2=0).
__device__ __forceinline__ void tdm_load_rows(uint64_t gaddr, uint32_t lds_addr) {
  u32x4 g0;
  g0[0] = 1u;                                              // count=1 (valid), no gather
  g0[1] = lds_addr;                                        // LDS byte address
  g0[2] = (uint32_t)gaddr;                                 // global_addr[31:0]
  g0[3] = (uint32_t)((gaddr >> 32) & 0x1FFFFFFu)           // global_addr[56:32]
          | (2u << 30);                                    // type = 2 ("image")
  i32x8 g1;
  g1[0] = (int)(2u << 16);                                 // data_size=2 -> 4 bytes
  g1[1] = (int)((uint32_t)DET << 16);                      // tensor_dim0 lo16 (=512)
  g1[2] = (int)((uint32_t)KA << 16);                       // dim0 hi16=0 | tensor_dim1 lo16 (=8)
  g1[3] = (int)((uint32_t)DET << 16);                      // dim1 hi16=0 | tile_dim0 (=512)
  g1[4] = (int)KA;                                         // tile_dim1 (=8) | tile_dim2=0
  g1[5] = (int)DET;                                        // tensor_dim0_stride lo32 (=512 elems)
  g1[6] = 0;                                               // stride0 hi16 | stride1 lo16
  g1[7] = 0;                                               // stride1 hi32
  i32x4 z4 = {0, 0, 0, 0};
  i32x8 z8 = {0, 0, 0, 0, 0, 0, 0, 0};
  __builtin_amdgcn_tensor_load_to_lds(g0, g1, z4, z4, z8, 0);
}
#endif

__global__ __launch_bounds__(THREADS) void AdjointOperator_backproject_kernel(
    const float* __restrict__ y,       // [B, 720, 512]
    const float* __restrict__ angles,  // [720] degrees
    float* __restrict__ out)           // [B, 512, 512]
{
  __shared__ float sC[NANG];
  __shared__ float sS[NANG];
  __shared__ float sRows[2][KA][DET];  // double-buffered angle-block staging

  const int tid = threadIdx.x;
  const int tx = tid & (TILE_X - 1);
  const int ty = tid >> 6;             // 0..3
  const int bx = blockIdx.x;
  const int by = blockIdx.y;
  const int b  = blockIdx.z;

  // Precompute rotation cos/sin for all angles (rad = -deg2rad(theta)).
  for (int a = tid; a < NANG; a += THREADS) {
    float rad = -angles[a] * 0.017453292519943295f;
    float sv, cv;
    __sincosf(rad, &sv, &cv);
    sC[a] = cv;
    sS[a] = sv;
  }

  const int px = bx * TILE_X + tx;
  const float u = (float)px - 255.5f;
  float v[RPT];
  float acc[RPT];
#pragma unroll
  for (int r = 0; r < RPT; ++r) {
    v[r] = (float)(by * TILE_Y + ty + 4 * r) - 255.5f;
    acc[r] = 0.0f;
  }

  const float* ybase = y + (size_t)b * NANG * DET;

#if USE_TDM
  const uint32_t ldsRows = (uint32_t)(uintptr_t)(&sRows[0][0][0]);
  const uint64_t gbase = (uint64_t)(uintptr_t)ybase;
  // Prologue: start DMA of block 0 (wave 0 only; TDM ignores EXEC).
  if (tid < 32) {
    tdm_load_rows(gbase, ldsRows);
  }
#endif

  for (int blk = 0; blk < NBLK; ++blk) {
#if USE_TDM
    if (tid < 32) {
      if (blk + 1 < NBLK) {
        // Pipeline: kick off next block, then wait until only it is in flight
        // (TDM ops from one wave complete in order => block `blk` has landed).
        tdm_load_rows(gbase + (uint64_t)(blk + 1) * KA * DET * sizeof(float),
                      ldsRows + (uint32_t)(((blk + 1) & 1) * KA * DET * sizeof(float)));
        __builtin_amdgcn_s_wait_tensorcnt(1);
      } else {
        __builtin_amdgcn_s_wait_tensorcnt(0);
      }
    }
    __syncthreads();                       // data of block `blk` visible to all waves
    const float(*rows)[DET] = sRows[blk & 1];
#else
    __syncthreads();                       // previous block fully consumed
    {
      const float4* src = (const float4*)(ybase + (size_t)blk * KA * DET);
      float4* dst = (float4*)(&sRows[0][0][0]);
      for (int i = tid; i < KA * DET / 4; i += THREADS) dst[i] = src[i];
    }
    __syncthreads();
    const float(*rows)[DET] = sRows[0];
#endif

    for (int a = 0; a < KA; ++a) {
      const int ga = blk * KA + a;
      const float c = sC[ga];
      const float sn = sS[ga];
      const float cu255 = fmaf(c, u, 255.5f);   // ix = cu255 - sn*v
      const float su255 = fmaf(sn, u, 255.5f);  // iy = su255 + c*v
      const float* __restrict__ row = rows[a];
#pragma unroll
      for (int r = 0; r < RPT; ++r) {
        float iy = fmaf(c, v[r], su255);
        float wy = fminf(iy + 1.0f, 512.0f - iy);
        wy = fminf(fmaxf(wy, 0.0f), 1.0f);

        float ix = fmaf(-sn, v[r], cu255);
        float x0f = floorf(ix);
        float f = ix - x0f;
        int i0 = (int)x0f;
        int i1 = i0 + 1;
        float w0 = ((unsigned)i0 < 512u) ? (1.0f - f) : 0.0f;
        float w1 = ((unsigned)i1 < 512u) ? f : 0.0f;
        int c0 = min(max(i0, 0), DET - 1);
        int c1 = min(max(i1, 0), DET - 1);
        float val = fmaf(w1, row[c1], w0 * row[c0]);
        acc[r] = fmaf(wy, val, acc[r]);
      }
    }
#if USE_TDM
    __syncthreads();   // everyone done with buf[blk&1] before it is reloaded
#endif
  }

  const float scale = 1.0f / (float)NANG;
  float* outB = out + (size_t)b * DET * DET;
#pragma unroll
  for (int r = 0; r < RPT; ++r) {
    const int py = by * TILE_Y + ty + 4 * r;
    outB[(size_t)py * DET + px] = acc[r] * scale;
  }
}

extern "C" void kernel_launch(void* const* d_in, const int* in_sizes, int n_in,
                              void* d_out, int out_size, void* d_ws, size_t ws_size,
                              hipStream_t stream) {
  (void)in_sizes; (void)n_in; (void)d_ws; (void)ws_size;
  const float* yin = (const float*)d_in[0];       // [B,1,720,512] f32
  const float* angles = (const float*)d_in[1];    // [720] f32
  float* out = (float*)d_out;                     // [B,1,512,512] f32
  const int B = out_size / (DET * DET);
  dim3 grid(DET / TILE_X, DET / TILE_Y, B);
  AdjointOperator_backproject_kernel<<<grid, THREADS, 0, stream>>>(yin, angles, out);
}